// RoutingByAggreement_41154376630984
// MI455X (gfx1250) — compile-verified
//
#include <hip/hip_runtime.h>
#include <stdint.h>

// Problem constants (from reference): u_hat (B, I, O, D) fp32
#define B_   256
#define I_   1152
#define O_   10
#define D_   16
#define ROWP 17          // LDS row pitch in dwords (16 data + 1 TDM pad -> bank-conflict free)
#define NITER 3
#define NTHREADS 256
#define NWAVES 8

typedef __attribute__((ext_vector_type(2))) float        v2f;
typedef __attribute__((ext_vector_type(8))) float        v8f;
typedef __attribute__((ext_vector_type(4))) unsigned int v4u;
typedef __attribute__((ext_vector_type(8))) int          v8i;
typedef __attribute__((ext_vector_type(4))) int          v4i;

__device__ __forceinline__ float wave_reduce_max(float v) {
  v = fmaxf(v, __shfl_xor(v, 16, 32));
  v = fmaxf(v, __shfl_xor(v, 8, 32));
  v = fmaxf(v, __shfl_xor(v, 4, 32));
  v = fmaxf(v, __shfl_xor(v, 2, 32));
  v = fmaxf(v, __shfl_xor(v, 1, 32));
  return v;
}
__device__ __forceinline__ float wave_reduce_sum(float v) {
  v += __shfl_xor(v, 16, 32);
  v += __shfl_xor(v, 8, 32);
  v += __shfl_xor(v, 4, 32);
  v += __shfl_xor(v, 2, 32);
  v += __shfl_xor(v, 1, 32);
  return v;
}

// One workgroup per (b, o): stage the 1152x16 u_hat slice into LDS once via the
// Tensor Data Mover, then run all 3 routing iterations entirely out of LDS.
// 2560 workgroups; u_hat (189 MB) is read from HBM exactly once (~8 us @ 23.3 TB/s).
__global__ __launch_bounds__(NTHREADS)
void caps_routing_kernel(const float* __restrict__ u_hat, float* __restrict__ out) {
  extern __shared__ float smem[];
  float* tile   = smem;                   // I_*ROWP = 19584 floats (padded u_hat slice)
  float* blogit = tile + I_ * ROWP;       // 1152 routing logits
  float* cbuf   = blogit + I_;            // 1152 exp(b - m)
  float* part   = cbuf + I_;              // 8 waves x 16 partial s[d]
  float* vV     = part + NWAVES * D_;     // 16: squashed v
  float* red    = vV + D_;                // 8: cross-wave reduce scratch
  float* sc     = red + NWAVES;           // 2: [0]=max, [1]=1/Z

  const int tid  = threadIdx.x;
  const int lane = tid & 31;
  const int wid  = tid >> 5;
  const int o    = blockIdx.x;
  const int b    = blockIdx.y;

  // zero the routing logits while the TDM fills the tile
  for (int i = tid; i < I_; i += NTHREADS) blogit[i] = 0.0f;

  // ---- TDM bulk load: 2-D tensor, tile 16 x 1152, global row stride 160 elems,
  //      pad 1 dword after every 16 dwords -> LDS row pitch of 17 dwords.
  if (tid == 0) {
    unsigned long long gaddr =
        (unsigned long long)(uintptr_t)(u_hat + ((size_t)b * I_ * O_ + (size_t)o) * D_);
    unsigned ldsa = (unsigned)(uintptr_t)tile;  // low 32 bits of flat shared addr = LDS byte addr

    v4u g0;
    g0.x = 1u;                                            // count=1, user descriptor
    g0.y = ldsa;                                          // lds_addr (bytes)
    g0.z = (unsigned)(gaddr & 0xFFFFFFFFull);             // global_addr[31:0]
    g0.w = (unsigned)((gaddr >> 32) & 0x1FFFFFFu) | (2u << 30); // global_addr[56:32] | type=2

    v8i g1;
    g1[0] = (int)0x00D20000;        // wg_mask=0 | data_size=2(4B) | pad_en=1 | pad_interval=3(16dw) | pad_amount=0(1dw)
    g1[1] = (int)(16u << 16);       // atomic_barrier_addr=0 | tensor_dim0 lo16 = 16
    g1[2] = (int)((unsigned)I_ << 16); // tensor_dim0 hi16=0 | tensor_dim1 lo16 = 1152
    g1[3] = (int)(16u << 16);       // tensor_dim1 hi16=0 | tile_dim0 = 16
    g1[4] = (int)I_;                // tile_dim1 = 1152 | tile_dim2 = 0
    g1[5] = (int)(O_ * D_);         // tensor_dim0_stride lo32 = 160
    g1[6] = 0;                      // stride hi16 | tensor_dim1_stride lo16 (unused, 2D)
    g1[7] = 0;

    v4i g2 = {0, 0, 0, 0};          // group 2 unused (2-D tensor)
    v4i g3 = {0, 0, 0, 0};          // group 3 unused (2-D tensor)
    v8i g4 = {0, 0, 0, 0, 0, 0, 0, 0}; // 6-arg toolchain form: extra descriptor block, unused
    __builtin_amdgcn_tensor_load_to_lds(g0, g1, g2, g3, g4, 0);
    __builtin_amdgcn_s_wait_tensorcnt(0);
  }
  __syncthreads();

  for (int r = 0; r < NITER; ++r) {
    // ---- 1) m = max_i logit[i]  (for stable softmax)
    float lm = -3.402823466e38f;
    for (int i = tid; i < I_; i += NTHREADS) lm = fmaxf(lm, blogit[i]);
    lm = wave_reduce_max(lm);
    if (lane == 0) red[wid] = lm;
    __syncthreads();
    if (wid == 0) {
      float t = (lane < NWAVES) ? red[lane] : -3.402823466e38f;
      t = wave_reduce_max(t);
      if (lane == 0) sc[0] = t;
    }
    __syncthreads();
    const float m = sc[0];

    // ---- 2) cbuf[i] = exp(logit[i] - m), Z = sum
    float lz = 0.0f;
    for (int i = tid; i < I_; i += NTHREADS) {
      float e = __expf(blogit[i] - m);
      cbuf[i] = e;
      lz += e;
    }
    lz = wave_reduce_sum(lz);
    if (lane == 0) red[wid] = lz;
    __syncthreads();
    if (wid == 0) {
      float t = (lane < NWAVES) ? red[lane] : 0.0f;
      t = wave_reduce_sum(t);
      if (lane == 0) sc[1] = 1.0f / t;
    }
    __syncthreads();
    const float invZ = sc[1];

    // ---- 3) s[d] = invZ * sum_i cbuf[i] * U[i,d]
    //      lane layout: d = tid&15, row-phase = tid>>4; 17-dword pitch -> conflict-free
    {
      const int d  = tid & 15;
      const int r0 = tid >> 4;            // 0..15
      float acc = 0.0f;
      for (int i = r0; i < I_; i += 16)
        acc = fmaf(cbuf[i], tile[i * ROWP + d], acc);
      acc += __shfl_xor(acc, 16, 32);     // lanes L and L+16 share d
      if (lane < 16) part[wid * D_ + d] = acc;   // deterministic 2-stage reduce
    }
    __syncthreads();

    // ---- 4) squash: v = s * ||s|| / (1 + ||s||^2)   (wave 0)
    if (wid == 0 && lane < D_) {
      float sd = 0.0f;
#pragma unroll
      for (int w = 0; w < NWAVES; ++w) sd += part[w * D_ + lane];
      sd *= invZ;
      float sq = sd * sd;
      sq += __shfl_xor(sq, 1, 32);
      sq += __shfl_xor(sq, 2, 32);
      sq += __shfl_xor(sq, 4, 32);
      sq += __shfl_xor(sq, 8, 32);
      float nrm = sqrtf(sq);
      float vd  = sd * (nrm / (1.0f + sq));
      vV[lane] = vd;
      if (r == NITER - 1) out[((size_t)b * O_ + o) * D_ + lane] = vd;
    }
    __syncthreads();

    // ---- 5) agreement: logit[i] += dot(U[i,:], v) via V_WMMA_F32_16X16X4_F32.
    //      A[m,k] = v[d0+k] (broadcast over rows), B[k,n] = U[ibase+n, d0+k];
    //      after 4 accumulating WMMAs, acc[0] at lane L = dot(U[ibase+(L&15)], v).
    //      A-fragments are loop-invariant -> hoisted into registers so the inner
    //      loop does only the 4 B-fragment ds_load_2addr pairs per group.
    if (r < NITER - 1) {
      const int kk = (lane >> 4) << 1;    // K-split per half-wave: 0 or 2
      const int rr = lane & 15;           // N index (row of U)
      const v2f A0 = {vV[0 + kk],  vV[1 + kk]};
      const v2f A1 = {vV[4 + kk],  vV[5 + kk]};
      const v2f A2 = {vV[8 + kk],  vV[9 + kk]};
      const v2f A3 = {vV[12 + kk], vV[13 + kk]};
      for (int g = wid; g < I_ / 16; g += NWAVES) {   // 72 groups, 9 per wave
        const int ibase = g * 16;
        const float* rowp = &tile[(ibase + rr) * ROWP + kk];
        const v2f B0 = {rowp[0],  rowp[1]};
        const v2f B1 = {rowp[4],  rowp[5]};
        const v2f B2 = {rowp[8],  rowp[9]};
        const v2f B3 = {rowp[12], rowp[13]};
        v8f acc = {0.f, 0.f, 0.f, 0.f, 0.f, 0.f, 0.f, 0.f};
        acc = __builtin_amdgcn_wmma_f32_16x16x4_f32(false, A0, false, B0, (short)0, acc, false, false);
        acc = __builtin_amdgcn_wmma_f32_16x16x4_f32(false, A1, false, B1, (short)0, acc, false, false);
        acc = __builtin_amdgcn_wmma_f32_16x16x4_f32(false, A2, false, B2, (short)0, acc, false, false);
        acc = __builtin_amdgcn_wmma_f32_16x16x4_f32(false, A3, false, B3, (short)0, acc, false, false);
        if (lane < 16) blogit[ibase + lane] += acc[0];
      }
    }
    __syncthreads();
  }
}

extern "C" void kernel_launch(void* const* d_in, const int* in_sizes, int n_in,
                              void* d_out, int out_size, void* d_ws, size_t ws_size,
                              hipStream_t stream) {
  const float* u_hat = (const float*)d_in[0];
  float* out = (float*)d_out;
  (void)in_sizes; (void)n_in; (void)out_size; (void)d_ws; (void)ws_size;

  // LDS: tile 19584 + blogit 1152 + cbuf 1152 + part 128 + vV 16 + red 8 + sc 2 = 22042 floats
  size_t shmem = (size_t)(I_ * ROWP + 2 * I_ + NWAVES * D_ + D_ + NWAVES + 2) * sizeof(float);
  dim3 grid(O_, B_);
  caps_routing_kernel<<<grid, dim3(NTHREADS), shmem, stream>>>(u_hat, out);
}